// EarlyFusionConv_54065048322100
// MI455X (gfx1250) — compile-verified
//
#include <hip/hip_runtime.h>

typedef __attribute__((ext_vector_type(2))) float v2f;
typedef __attribute__((ext_vector_type(4))) float v4f;
typedef __attribute__((ext_vector_type(8))) float v8f;

#define BN_EPS 1e-5f

// ---------------------------------------------------------------------------
// Kernel 1: one wave (32 lanes) computes act[4][32] =
//   relu(BN( (tab @ Wp^T + bp) @ Wc^T ))  using fp32 WMMA 16x16x4.
//
// WMMA f32 16x16x4 layouts (ISA 7.12.2):
//   A (16x4, MxK): lane l<16  -> {A[M=l][K=0], A[M=l][K=1]}
//                  lane l>=16 -> {A[M=l-16][K=2], A[M=l-16][K=3]}
//   B (4x16, KxN): same split, N striped across lanes within a VGPR.
//   C/D (16x16):   lane l<16: vgpr r = D[M=r][N=l]; lanes 16..31: M=r+8.
// M=4 padded to 16 with zeros; N=32 handled as two 16-wide tiles.
// ---------------------------------------------------------------------------
__global__ __launch_bounds__(32) void head_kernel(
    const float* __restrict__ tab,   // (4, 64)
    const float* __restrict__ Wp,    // (32, 64)
    const float* __restrict__ bp,    // (32,)
    const float* __restrict__ Wc,    // (32, 32)
    const float* __restrict__ gamma, // (32,)
    const float* __restrict__ beta,  // (32,)
    float* __restrict__ act)         // (4, 32) out, in workspace
{
    __shared__ float projLDS[16 * 32]; // proj padded to 16 rows, row stride 32

    const int lane = threadIdx.x;
    const int half = lane >> 4;  // 0 -> K pair {0,1}, 1 -> K pair {2,3}
    const int l    = lane & 15;  // M index (A side) / N index (B side)

    // Zero-pad the proj staging buffer (rows 4..15 must be zero).
    for (int i = lane; i < 16 * 32; i += 32) projLDS[i] = 0.0f;
    __syncthreads();

    // ---- GEMM 1: proj(16x32) = tab(16x64, rows 4..15 zero) @ Wp^T(64x32) ----
    v8f acc0 = {}; v8f acc1 = {};
#pragma unroll
    for (int k0 = 0; k0 < 64; k0 += 4) {
        const int ka = k0 + 2 * half;
        v2f a;
        a.x = (l < 4) ? tab[l * 64 + ka + 0] : 0.0f;
        a.y = (l < 4) ? tab[l * 64 + ka + 1] : 0.0f;
        v2f b0; b0.x = Wp[l * 64 + ka];          b0.y = Wp[l * 64 + ka + 1];
        v2f b1; b1.x = Wp[(16 + l) * 64 + ka];   b1.y = Wp[(16 + l) * 64 + ka + 1];
        acc0 = __builtin_amdgcn_wmma_f32_16x16x4_f32(
            false, a, false, b0, (short)0, acc0, false, false);
        acc1 = __builtin_amdgcn_wmma_f32_16x16x4_f32(
            false, a, false, b1, (short)0, acc1, false, false);
    }

    // Rows 0..3 live in lanes 0..15, acc vgprs 0..3. Add bias, stage to LDS.
    if (half == 0) {
#pragma unroll
        for (int r = 0; r < 4; ++r) {
            projLDS[r * 32 + l]      = acc0[r] + bp[l];
            projLDS[r * 32 + 16 + l] = acc1[r] + bp[16 + l];
        }
    }
    __syncthreads();

    // ---- GEMM 2: mixed(16x32) = proj(16x32) @ Wc^T(32x32) ----
    v8f mv0 = {}; v8f mv1 = {};
#pragma unroll
    for (int k0 = 0; k0 < 32; k0 += 4) {
        const int ka = k0 + 2 * half;
        v2f a;  a.x  = projLDS[l * 32 + ka];     a.y  = projLDS[l * 32 + ka + 1];
        v2f b0; b0.x = Wc[l * 32 + ka];          b0.y = Wc[l * 32 + ka + 1];
        v2f b1; b1.x = Wc[(16 + l) * 32 + ka];   b1.y = Wc[(16 + l) * 32 + ka + 1];
        mv0 = __builtin_amdgcn_wmma_f32_16x16x4_f32(
            false, a, false, b0, (short)0, mv0, false, false);
        mv1 = __builtin_amdgcn_wmma_f32_16x16x4_f32(
            false, a, false, b1, (short)0, mv1, false, false);
    }

    // ---- BatchNorm over N=4 (per channel = per lane) + ReLU ----
    if (half == 0) {
#pragma unroll
        for (int t = 0; t < 2; ++t) {
            const v8f mv = t ? mv1 : mv0;
            const int c = 16 * t + l;
            const float x0 = mv[0], x1 = mv[1], x2 = mv[2], x3 = mv[3];
            const float mean = 0.25f * (x0 + x1 + x2 + x3);
            const float d0 = x0 - mean, d1 = x1 - mean,
                        d2 = x2 - mean, d3 = x3 - mean;
            const float var = 0.25f * (d0 * d0 + d1 * d1 + d2 * d2 + d3 * d3);
            const float sg = gamma[c] * rsqrtf(var + BN_EPS);
            const float bt = beta[c];
            act[0 * 32 + c] = fmaxf(d0 * sg + bt, 0.0f);
            act[1 * 32 + c] = fmaxf(d1 * sg + bt, 0.0f);
            act[2 * 32 + c] = fmaxf(d2 * sg + bt, 0.0f);
            act[3 * 32 + c] = fmaxf(d3 * sg + bt, 0.0f);
        }
    }
}

// ---------------------------------------------------------------------------
// Kernel 2: out(4, 64, 96^3): channels 0..31 copy volume, 32..63 broadcast
// act[n][c-32]. Pure streaming -> B128 non-temporal loads/stores, no integer
// division in the hot path (blockIdx.y encodes (n,c)).
// S4 = 96^3/4 = 221184 float4 per (n,c) slice = 216 blocks * 256 thr * 4.
// ---------------------------------------------------------------------------
__global__ __launch_bounds__(256) void bcast_concat_kernel(
    const float* __restrict__ vol,  // (4, 32, 96^3)
    const float* __restrict__ act,  // (4, 32)
    float* __restrict__ outp)       // (4, 64, 96^3)
{
    const int S4 = (96 * 96 * 96) / 4; // 221184 float4 per slice
    const int nc = blockIdx.y;         // 0..255
    const int n  = nc >> 6;
    const int c  = nc & 63;

    v4f* __restrict__ out4 = (v4f*)outp + (size_t)nc * S4;
    const int base = blockIdx.x * (256 * 4) + threadIdx.x;

    if (c < 32) {
        const v4f* __restrict__ src = (const v4f*)vol + (size_t)(n * 32 + c) * S4;
#pragma unroll
        for (int i = 0; i < 4; ++i) {
            const int p = base + i * 256;
            v4f v = __builtin_nontemporal_load(&src[p]);
            __builtin_nontemporal_store(v, &out4[p]);
        }
    } else {
        const float a = act[n * 32 + (c - 32)];
        v4f v = {a, a, a, a};
#pragma unroll
        for (int i = 0; i < 4; ++i)
            __builtin_nontemporal_store(v, &out4[base + i * 256]);
    }
}

extern "C" void kernel_launch(void* const* d_in, const int* in_sizes, int n_in,
                              void* d_out, int out_size, void* d_ws, size_t ws_size,
                              hipStream_t stream) {
    (void)in_sizes; (void)n_in; (void)out_size; (void)ws_size;
    const float* vol = (const float*)d_in[0]; // (4,32,96,96,96)
    const float* tab = (const float*)d_in[1]; // (4,64)
    const float* Wp  = (const float*)d_in[2]; // (32,64)
    const float* bp  = (const float*)d_in[3]; // (32,)
    const float* Wc  = (const float*)d_in[4]; // (32,32)
    const float* gm  = (const float*)d_in[5]; // (32,)
    const float* bt  = (const float*)d_in[6]; // (32,)

    float* act = (float*)d_ws; // 4*32 floats of scratch

    head_kernel<<<1, 32, 0, stream>>>(tab, Wp, bp, Wc, gm, bt, act);
    bcast_concat_kernel<<<dim3(216, 256), 256, 0, stream>>>(
        vol, act, (float*)d_out);
}